// SKipPool_Full_25890062861063
// MI455X (gfx1250) — compile-verified
//
#include <hip/hip_runtime.h>
#include <hip/hip_bf16.h>

typedef float v2f __attribute__((ext_vector_type(2)));
typedef float v8f __attribute__((ext_vector_type(8)));

namespace {
constexpr int Bn   = 64;
constexpr int Nn   = 512;
constexpr int Cn   = 128;
constexpr int FSn  = 16;
constexpr int RESn = 511;
constexpr int XS   = 132;   // xg LDS row stride (floats): 16B aligned, conflict-friendly
constexpr int BS   = 17;    // band row stride (floats): odd -> conflict-free row access
constexpr float EPSc = 1e-20f;
constexpr float NEGc = -1.0e9f;
// LDS layout (floats): xg | band | skey | sidx | nrm2 | asr | adt | wvs | jmp | swl | asl | adl
constexpr size_t SMEM_FLOATS = (size_t)Nn * XS + (size_t)Nn * BS + 7u * Nn + 3u * Cn;
constexpr size_t SMEM_BYTES  = SMEM_FLOATS * 4;   // = 321024 B  (<= 320 KB WGP LDS)
}

__global__ __launch_bounds__(Nn)
void skippool_kernel(const float* __restrict__ x,
                     const float* __restrict__ score_w,
                     const float* __restrict__ att_params,
                     const int*   __restrict__ epoch_p,
                     float*       __restrict__ out)
{
    extern __shared__ float smem[];
    float* xg   = smem;                    // 512*132
    float* band = xg + Nn * XS;            // 512*17  (Gram band G[r][j])
    float* skey = band + Nn * BS;          // 512 sorted scores
    int*   sidx = (int*)(skey + Nn);       // 512 permutation
    float* nrm2 = (float*)(sidx + Nn);     // 512 row squared norms
    float* asr  = nrm2 + Nn;               // 512 xg . a_src
    float* adt  = asr + Nn;                // 512 xg . a_dst
    float* wvs  = adt + Nn;                // 512 w_row * tanh(score)
    int*   jmp  = (int*)(wvs + Nn);        // 512 jumps
    float* swl  = (float*)(jmp + Nn);      // 128 score_w
    float* asl  = swl + Cn;                // 128 a_src
    float* adl  = asl + Cn;                // 128 a_dst

    const int tid = threadIdx.x;
    const int b   = blockIdx.x;

    // ---- stage weights ----
    if (tid < Cn) {
        swl[tid] = score_w[tid];
        asl[tid] = att_params[tid];
        adl[tid] = att_params[Cn + tid];
    }
    __syncthreads();

    // ---- scores: (x . w) / ||w|| ----
    {
        const float* xrow = x + (size_t)(b * Nn + tid) * Cn;
        float dot = 0.f, nw = 0.f;
        #pragma unroll 8
        for (int c = 0; c < Cn; ++c) {
            float w = swl[c];
            dot += xrow[c] * w;
            nw  += w * w;
        }
        skey[tid] = dot * rsqrtf(nw);
        sidx[tid] = tid;
    }

    // ---- bitonic sort, descending (key, idx) ----
    for (int k = 2; k <= Nn; k <<= 1) {
        for (int j = k >> 1; j > 0; j >>= 1) {
            __syncthreads();
            int ixj = tid ^ j;
            if (ixj > tid) {
                float a = skey[tid], c = skey[ixj];
                bool up = (tid & k) == 0;
                if (up ? (a < c) : (a > c)) {
                    skey[tid] = c; skey[ixj] = a;
                    int t0 = sidx[tid]; sidx[tid] = sidx[ixj]; sidx[ixj] = t0;
                }
            }
        }
    }
    __syncthreads();

    // ---- gather sorted rows into LDS (512B/row/thread, b128 stores) ----
    {
        const float4* src = (const float4*)(x + (size_t)(b * Nn + sidx[tid]) * Cn);
        float4* dst = (float4*)(xg + tid * XS);
        #pragma unroll
        for (int c4 = 0; c4 < Cn / 4; ++c4) dst[c4] = src[c4];
    }
    __syncthreads();

    // ---- per-row reductions: ||row||^2, row.a_src, row.a_dst ----
    {
        const float* row = xg + tid * XS;
        float n2 = 0.f, s0 = 0.f, s1 = 0.f;
        #pragma unroll 8
        for (int c = 0; c < Cn; ++c) {
            float v = row[c];
            n2 += v * v;
            s0 += v * asl[c];
            s1 += v * adl[c];
        }
        nrm2[tid] = n2; asr[tid] = s0; adt[tid] = s1;
    }
    __syncthreads();

    // ---- banded Gram via V_WMMA_F32_16X16X4_F32 (diag + superdiag 16x16 tiles) ----
    {
        const int wave = tid >> 5;
        const int lane = tid & 31;
        const int half = lane >> 4;     // K-pair select (ISA 16x4 f32 A layout)
        const int mn   = lane & 15;     // M for A, N for B
        for (int tt = wave; tt < 63; tt += 16) {      // tile 63 (i=31,t=1) is out of range
            int i = tt >> 1, t = tt & 1;
            int rowbase = i * 16;
            int colbase = rowbase + t * 16;
            const float* arow = xg + (rowbase + mn) * XS + half * 2;
            const float* brow = xg + (colbase + mn) * XS + half * 2;  // B[k][n]=xg[col+n][k]
            v8f acc = {};
            #pragma unroll
            for (int kk = 0; kk < Cn; kk += 4) {
                v2f a  = *(const v2f*)(arow + kk);
                v2f bb = *(const v2f*)(brow + kk);
                acc = __builtin_amdgcn_wmma_f32_16x16x4_f32(
                          false, a, false, bb, (short)0, acc, false, false);
            }
            #pragma unroll
            for (int v = 0; v < 8; ++v) {           // C/D: M=v+8*half, N=mn
                int rr = rowbase + v + half * 8;
                int jj = colbase + mn - rr;
                if (rr < RESn && jj >= 0 && jj < FSn)
                    band[rr * BS + jj] = acc[v];
            }
        }
    }
    __syncthreads();

    // ---- per-row: distances, attention softmax, w_row, jumps ----
    {
        int ep = *epoch_p;
        float tau0 = 10.0f * powf(0.01f, (float)ep * 0.01f);
        if (tid < RESn) {
            float myn = nrm2[tid];
            float as  = asr[tid];
            float att[FSn], kv[FSn];
            float amax = -3.0e38f;
            #pragma unroll
            for (int j = 0; j < FSn; ++j) {
                int dest = tid + j;
                if (dest < Nn) {
                    float g  = band[tid * BS + j];
                    float d2 = fmaxf(myn + nrm2[dest] - 2.0f * g, 0.0f);
                    kv[j]  = __expf(-0.5f * sqrtf(d2)) - EPSc;
                    att[j] = as + adt[dest];
                } else {
                    kv[j]  = -EPSc;
                    att[j] = NEGc;
                }
                amax = fmaxf(amax, att[j]);
            }
            float itau = 1.0f / tau0;
            float ssum = 0.f, w[FSn];
            #pragma unroll
            for (int j = 0; j < FSn; ++j) {
                float e = __expf((att[j] - amax) * itau);
                w[j] = e; ssum += e;
            }
            float inv = 1.0f / ssum;
            float wmaxv = -3.0e38f, best = -3.0e38f;
            int bj = 1;
            #pragma unroll
            for (int j = 0; j < FSn; ++j) {
                float q  = kv[j] * w[j] * inv;
                float qm = (tid + j < Nn) ? q : NEGc;
                wmaxv = fmaxf(wmaxv, qm);
                if (j >= 1 && qm > best) { best = qm; bj = j; }   // first-max, like jnp.argmax
            }
            wvs[tid] = wmaxv * tanhf(skey[tid]);
            jmp[tid] = bj;
        }
    }
    __syncthreads();

    // ---- sequential skip-chain (wave 0), 4 channels per lane ----
    if (tid < 32) {
        float a0 = 0.f, a1 = 0.f, a2 = 0.f, a3 = 0.f;
        int p = 0;
        while (p < RESn) {
            float wv = wvs[p];
            const float* row = xg + p * XS;
            a0 += wv * row[tid];
            a1 += wv * row[tid + 32];
            a2 += wv * row[tid + 64];
            a3 += wv * row[tid + 96];
            p += jmp[p];
        }
        float* o = out + b * Cn;
        o[tid] = a0; o[tid + 32] = a1; o[tid + 64] = a2; o[tid + 96] = a3;
    }
}

extern "C" void kernel_launch(void* const* d_in, const int* in_sizes, int n_in,
                              void* d_out, int out_size, void* d_ws, size_t ws_size,
                              hipStream_t stream) {
    const float* x          = (const float*)d_in[0];
    /* d_in[1] = edge_index: unused by the reference forward pass */
    const float* score_w    = (const float*)d_in[2];
    const float* att_params = (const float*)d_in[3];
    const int*   epoch      = (const int*)d_in[4];
    float* out = (float*)d_out;

    skippool_kernel<<<dim3(Bn), dim3(Nn), SMEM_BYTES, stream>>>(
        x, score_w, att_params, epoch, out);
}